// Direction_24773371363600
// MI455X (gfx1250) — compile-verified
//
#include <hip/hip_runtime.h>

// ---------------------------------------------------------------------------
// Problem: out[b, o] = sum_m input[b, m] * Q[o, m]
//   input : [524288, 20]  f32
//   weight: [512, 20]     f32   -> Q = qr(weight + 1e-8).Q  (512 x 20)
//   out   : [524288, 512] f32
//
// Store-bandwidth bound (1.07 GB out @ 23.3 TB/s ~= 48 us). Use fp32 WMMA
// V_WMMA_F32_16X16X4_F32: K=20 = 5 x 4, full fp32 accuracy.
// ---------------------------------------------------------------------------

typedef __attribute__((ext_vector_type(2))) float v2f;
typedef __attribute__((ext_vector_type(8))) float v8f;

#define QR_ROWS   512
#define QR_COLS   20
#define OUT_N     512
#define LDSQ_STR  22      // padded LDS row stride: even (8B-aligned b64 loads)
                          // and 22*l mod 64 distinct over 32 lanes -> no bank conflicts

// ------------------------- Kernel 1: QR (MGS) -------------------------------
// One block, 512 threads; thread i owns row i of W (20 floats in registers).
// Modified Gram-Schmidt, batched reductions: at step j compute raw dots
// r_k = w_j . w_k for k = j..19 in one LDS round, then
//   q_j = w_j * rsqrt(r_j);   w_k -= (r_k / r_j) * w_j.
__global__ __launch_bounds__(512) void qr_mgs_kernel(const float* __restrict__ W,
                                                     float* __restrict__ Q) {
  __shared__ float part[16][QR_COLS];   // per-wave partial dots
  __shared__ float dots[QR_COLS];

  const int tid  = threadIdx.x;         // row index 0..511
  const int lane = tid & 31;
  const int wid  = tid >> 5;            // 16 waves

  float w[QR_COLS];
#pragma unroll
  for (int c = 0; c < QR_COLS; ++c)
    w[c] = W[tid * QR_COLS + c] + 1e-8f;

  for (int j = 0; j < QR_COLS; ++j) {
    // wave-level partial dots w_j . w_k  (k = j..19)
    for (int k = j; k < QR_COLS; ++k) {
      float v = w[j] * w[k];
#pragma unroll
      for (int o = 16; o > 0; o >>= 1)
        v += __shfl_down(v, o, 32);
      if (lane == 0) part[wid][k] = v;
    }
    __syncthreads();
    // fold 16 wave partials; thread 'k' owns dot k
    if (tid >= j && tid < QR_COLS) {
      float s = 0.f;
#pragma unroll
      for (int ww = 0; ww < 16; ++ww) s += part[ww][tid];
      dots[tid] = s;
    }
    __syncthreads();

    const float rjj  = dots[j];
    const float inv  = rsqrtf(rjj);
    const float wjn  = 1.0f / rjj;
    const float wj   = w[j];
    w[j] = wj * inv;                       // q_j
    for (int k = j + 1; k < QR_COLS; ++k)
      w[k] -= dots[k] * wjn * wj;          // orthogonalize remaining columns
    // no extra sync needed: 'part' is only read after next __syncthreads(),
    // 'dots' only rewritten after the barrier everyone must pass.
  }

#pragma unroll
  for (int c = 0; c < QR_COLS; ++c)
    Q[tid * QR_COLS + c] = w[c];
}

// ------------------------- Kernel 2: WMMA GEMM ------------------------------
// Block = 256 threads (8 waves). Block covers 128 rows x all 512 cols.
// Each wave: 16 rows, loops 32 N-tiles of 16 cols; 5 chained
// v_wmma_f32_16x16x4_f32 per tile (K = 20).
__global__ __launch_bounds__(256) void gemm_wmma_kernel(const float* __restrict__ X,
                                                        const float* __restrict__ Q,
                                                        float* __restrict__ out) {
  __shared__ float ldsq[QR_ROWS * LDSQ_STR];   // 44 KB

  const int tid = threadIdx.x;
  // Stage Q (512 x 20) into LDS with padded stride, coalesced.
  for (int i = tid; i < QR_ROWS * QR_COLS; i += 256) {
    const int r = i / QR_COLS;
    const int c = i - r * QR_COLS;
    ldsq[r * LDSQ_STR + c] = Q[i];
  }
  __syncthreads();

  const int lane = tid & 31;
  const int wave = tid >> 5;
  const int half = lane >> 4;      // 0: K pair {0,1}, 1: K pair {2,3}
  const int l16  = lane & 15;

  const long m0   = (long)blockIdx.x * 128 + (long)wave * 16;
  const long arow = m0 + l16;

  // A fragments (ISA 32-bit A 16x4 layout): lane<16 -> K0,K1 ; lane>=16 -> K2,K3
  v2f a[5];
  const float* xrow = X + arow * QR_COLS + half * 2;
#pragma unroll
  for (int kk = 0; kk < 5; ++kk)
    a[kk] = *(const v2f*)(xrow + kk * 4);     // 8B-aligned (even float offset)

  for (int nt = 0; nt < 32; ++nt) {
    const int n0 = nt * 16;
    const float* qcol = &ldsq[(n0 + l16) * LDSQ_STR + half * 2];

    v8f acc = {0.f, 0.f, 0.f, 0.f, 0.f, 0.f, 0.f, 0.f};
#pragma unroll
    for (int kk = 0; kk < 5; ++kk) {
      // B fragment: B[k][n] = Q[n0+l16][4*kk + 2*half + {0,1}]
      const v2f b = *(const v2f*)(qcol + kk * 4);
      acc = __builtin_amdgcn_wmma_f32_16x16x4_f32(
          /*neg_a=*/false, a[kk], /*neg_b=*/false, b,
          /*c_mod=*/(short)0, acc, /*reuse_a=*/false, /*reuse_b=*/false);
    }

    // D layout: VGPR r -> row m0 + half*8 + r, col n0 + l16.
    // Each half-wave writes a contiguous 64B row segment per store.
    float* obase = out + (m0 + half * 8) * (long)OUT_N + n0 + l16;
#pragma unroll
    for (int r = 0; r < 8; ++r)
      obase[(long)r * OUT_N] = acc[r];
  }
}

// ----------------------------- Launcher -------------------------------------
extern "C" void kernel_launch(void* const* d_in, const int* in_sizes, int n_in,
                              void* d_out, int out_size, void* d_ws, size_t ws_size,
                              hipStream_t stream) {
  const float* X = (const float*)d_in[0];   // [batch, 20]
  const float* W = (const float*)d_in[1];   // [512, 20]
  float* out = (float*)d_out;               // [batch, 512]
  float* Qws = (float*)d_ws;                // 512*20 floats scratch

  const int batch = in_sizes[0] / QR_COLS;  // 524288
  const int grid  = batch / 128;            // 4096 (exact)

  qr_mgs_kernel<<<1, 512, 0, stream>>>(W, Qws);
  gemm_wmma_kernel<<<grid, 256, 0, stream>>>(X, Qws, out);
}